// PatchEmbedding_22754736734988
// MI455X (gfx1250) — compile-verified
//
#include <hip/hip_runtime.h>
#include <hip/hip_bf16.h>

// PatchEmbedding as GEMM: O[768,4096] = W[768,3072] x P[3072,4096] + b.
// CDNA5 paths: TDM tensor_load_to_lds stages both tiles (the patch tile is
// two contiguous rows of x because a 32-aligned K-block is one (c,py) plane),
// s_wait_tensorcnt + barriers for ordering, V_WMMA_F32_16X16X4_F32 compute.

typedef __attribute__((ext_vector_type(2))) float        v2f;
typedef __attribute__((ext_vector_type(8))) float        v8f;
typedef __attribute__((ext_vector_type(4))) unsigned int u32x4;
typedef __attribute__((ext_vector_type(8))) int          i32x8;
typedef __attribute__((ext_vector_type(4))) int          i32x4;

#define MTOT 768
#define NTOT 4096
#define KTOT 3072
#define BM   128
#define BN   128
#define KB   32
#define PAD  2          // 2 DWORDs of LDS padding per 32-DWORD row (TDM pad)
#define LDW  (KB + PAD) // 34-float LDS row: 16 lanes x stride 34 -> no bank conflicts

// Issue one 2D TDM tile load: tile (tile0 x tile1) of 4-byte elements from a
// row-major tensor (row length stride0 elements) into LDS at lds_off, with
// 2-DWORD padding inserted after every 32 DWORDs (rows become LDW floats).
__device__ __forceinline__ void tdm_load_2d(unsigned int lds_off, const void* gaddr,
                                            unsigned int tdim0, unsigned int tdim1,
                                            unsigned int tile0, unsigned int tile1,
                                            unsigned int stride0)
{
    unsigned long long ga = (unsigned long long)(uintptr_t)gaddr;
    u32x4 g0;
    g0.x = 1u;                                   // count=1 (valid user descriptor)
    g0.y = lds_off;                              // D#.lds_addr (bytes)
    g0.z = (unsigned int)ga;                     // global_addr[31:0]
    g0.w = (unsigned int)((ga >> 32) & 0x01FFFFFFu) | (2u << 30);  // addr[56:32] | type=2

    i32x8 g1;
    // data_size=4B (2), pad_enable, pad_interval=32 DW (code 4), pad_amount=2 DW (code 1)
    g1[0] = (int)((2u << 16) | (1u << 20) | (4u << 22) | (1u << 25));
    g1[1] = (int)((tdim0 & 0xFFFFu) << 16);                            // tensor_dim0[15:0]
    g1[2] = (int)(((tdim0 >> 16) & 0xFFFFu) | ((tdim1 & 0xFFFFu) << 16));
    g1[3] = (int)(((tdim1 >> 16) & 0xFFFFu) | (tile0 << 16));          // tile_dim0
    g1[4] = (int)(tile1 & 0xFFFFu);                                    // tile_dim1 (tile_dim2=0)
    g1[5] = (int)stride0;                                              // tensor_dim0_stride[31:0]
    g1[6] = 0;
    g1[7] = 0;

    i32x4 z4 = {0, 0, 0, 0};
#if __clang_major__ >= 23
    i32x8 z8 = {0, 0, 0, 0, 0, 0, 0, 0};
    __builtin_amdgcn_tensor_load_to_lds(g0, g1, z4, z4, z8, 0);
#else
    __builtin_amdgcn_tensor_load_to_lds(g0, g1, z4, z4, 0);
#endif
}

__global__ __launch_bounds__(256)
void patch_embed_wmma(const float* __restrict__ x,
                      const float* __restrict__ W,
                      const float* __restrict__ bias,
                      float* __restrict__ out)
{
    __shared__ float As[BM][LDW];   // W tile,     [m][k]
    __shared__ float Bs[BN][LDW];   // patch tile, [n][k] (memory order of x!)

    const int t    = threadIdx.x;   // 0..255 (8 wave32 waves)
    const int wave = t >> 5;
    const int lane = t & 31;
    const int half = lane >> 4;     // lanes 0-15 vs 16-31
    const int l16  = lane & 15;

    const int tileN0 = blockIdx.x * BN;   // gridDim.x = 32
    const int tileM0 = blockIdx.y * BM;   // gridDim.y = 6
    const int gy0    = tileN0 >> 6;       // first patch-grid row of this N tile

    // LDS aperture keeps the byte offset in the low 32 address bits.
    const unsigned int asOff = (unsigned int)(uintptr_t)&As[0][0];
    const unsigned int bsOff = (unsigned int)(uintptr_t)&Bs[0][0];

    v8f acc[8] = {};   // wave's 16 M-rows x 8 N-subtiles of 16

    for (int k0 = 0; k0 < KTOT; k0 += KB) {
        // A 32-aligned K-block is one (c, py) plane with px = 0..31.
        const int c  = k0 >> 10;
        const int py = (k0 & 1023) >> 5;

        if (t < 32) {  // wave 0 drives the Tensor Data Mover
            // A tile: 32 x 128 of W at (k0, tileM0); row stride KTOT.
            tdm_load_2d(asOff, &W[(unsigned)tileM0 * KTOT + k0],
                        KTOT, MTOT, KB, BM, KTOT);
            // B tile: two full contiguous 2048-float rows of x
            // (rows c*2048 + gy*32 + py for gy = gy0, gy0+1) -> Bs[n][k].
            tdm_load_2d(bsOff, &x[(unsigned)((c << 11) + (gy0 << 5) + py) * 2048],
                        2048, 3 * 2048, 2048, 2, 32 * 2048);
            __builtin_amdgcn_s_wait_tensorcnt(0);
        }
        __syncthreads();

        const int mRow = wave * 16 + l16;
        #pragma unroll
        for (int kk = 0; kk < KB; kk += 4) {
            // A 16x4 fragment: vgpr j holds K = kk + 2*half + j  -> one b64 load
            v2f a = *(const v2f*)&As[mRow][kk + 2 * half];
            #pragma unroll
            for (int nt = 0; nt < 8; ++nt) {
                // B 4x16 fragment: vgpr j holds K = kk + 2*half + j, N = l16
                v2f bf = *(const v2f*)&Bs[nt * 16 + l16][kk + 2 * half];
                acc[nt] = __builtin_amdgcn_wmma_f32_16x16x4_f32(
                    false, a, false, bf, (short)0, acc[nt], false, false);
            }
        }
        __syncthreads();   // all reads done before TDM overwrites LDS
    }

    // Epilogue: C layout vgpr v, lane L -> M = v + 8*half, N = l16.
    const int mBase = tileM0 + wave * 16;
    #pragma unroll
    for (int nt = 0; nt < 8; ++nt) {
        int n = tileN0 + nt * 16 + l16;
        #pragma unroll
        for (int v = 0; v < 8; ++v) {
            int m = mBase + v + 8 * half;
            out[m * NTOT + n] = acc[nt][v] + bias[m];
        }
    }
}

extern "C" void kernel_launch(void* const* d_in, const int* in_sizes, int n_in,
                              void* d_out, int out_size, void* d_ws, size_t ws_size,
                              hipStream_t stream) {
    // setup_inputs order: x, importance_weights (dead), W, b, num_patches (dead)
    const float* x    = (const float*)d_in[0];
    const float* W    = (const float*)d_in[2];
    const float* bias = (const float*)d_in[3];
    float* out        = (float*)d_out;   // [1, 768, 4096] row-major

    dim3 grid(NTOT / BN, MTOT / BM);     // 32 x 6
    dim3 block(256);
    patch_embed_wmma<<<grid, block, 0, stream>>>(x, W, bias, out);
}